// LNN_1314259993018
// MI455X (gfx1250) — compile-verified
//
#include <hip/hip_runtime.h>
#include <math.h>

#define NDOF   8
#define DIN    16
#define DH     128
#define BATCH  4096
#define EPB    4            // elements per block = waves per block
#define PAD    4
#define LSTR   (DH + PAD)   // LDS row stride (132 words -> conflict-free fragments)

typedef __attribute__((ext_vector_type(2))) float v2f;
typedef __attribute__((ext_vector_type(4))) float v4f;
typedef __attribute__((ext_vector_type(8))) float v8f;

__device__ __forceinline__ float softplus_f(float x) {
    return (x > 20.0f) ? x : log1pf(__expf(x));
}
__device__ __forceinline__ float sigmoid_f(float x) {
    return 1.0f / (1.0f + __expf(-x));
}

// D = A(16x4) * B(4x16) + C ; f32 WMMA
__device__ __forceinline__ v8f wmma4(v2f a, v2f b, v8f c) {
    return __builtin_amdgcn_wmma_f32_16x16x4_f32(false, a, false, b, (short)0, c,
                                                 false, false);
}

// ---------------------------------------------------------------------------
// Prep kernel: build (a) fragment-major W2 for the Q-GEMM B operand and
// (b) W2^T for the a2 matvec, both in workspace. 16384 threads.
//   w2f[(f*8+t)*32 + L] = { W2[4f+koff][16t+hl], W2[4f+koff+1][16t+hl] }
//   w2t[k*128+j]        = W2[j][k]
// ---------------------------------------------------------------------------
__global__ __launch_bounds__(256) void lnn_prep(const float* __restrict__ w2,
                                                float2* __restrict__ w2f,
                                                float*  __restrict__ w2t)
{
    int idx = blockIdx.x * 256 + threadIdx.x;          // [0, 16384)
    // W2^T
    int j = idx & (DH - 1);
    int k = idx >> 7;
    w2t[idx] = w2[j * DH + k];
    // fragment-major W2 (first 8192 threads)
    if (idx < 8192) {
        int L  = idx & 31;
        int t  = (idx >> 5) & 7;
        int f  = idx >> 8;                             // k-step index, 0..31
        int koff = (L >> 4) << 1;
        int hl   = L & 15;
        int ka   = 4 * f + koff;
        float2 v;
        v.x = w2[(ka)     * DH + 16 * t + hl];
        v.y = w2[(ka + 1) * DH + 16 * t + hl];
        w2f[idx] = v;
    }
}

// ---------------------------------------------------------------------------
// Main kernel: one wave = one batch element.
// ---------------------------------------------------------------------------
__global__ __launch_bounds__(EPB * 32) void lnn_kernel(
    const float* __restrict__ z,  const float* __restrict__ w1,
    const float* __restrict__ b1, const float* __restrict__ w2,
    const float* __restrict__ b2, const float* __restrict__ w3,
    const float2* __restrict__ w2f, const float* __restrict__ w2t,
    int useWs, float* __restrict__ out)
{
    __shared__ float W1s[DIN][LSTR];              // 8.4 KB
    __shared__ float b1s[DH], b2s[DH], w3s[DH];   // 1.5 KB
    __shared__ float zs[EPB][DIN];
    __shared__ float hs[EPB][DH],  s1s[EPB][DH];
    __shared__ float g2s[EPB][DH], d2s[EPB][DH];
    __shared__ float d1s[EPB][DH], g1s[EPB][DH];  // 12 KB vectors
    __shared__ float jacs[EPB][DIN];
    __shared__ float Qs[EPB][DIN][LSTR];          // 33.8 KB (Q staging)
    __shared__ float Hs[EPB][NDOF][DIN];          // rows 8..15 of Hessian
    __shared__ float Aug[EPB][NDOF][NDOF + 1];

    const int tid  = threadIdx.x;
    const int w    = tid >> 5;
    const int lane = tid & 31;
    const int elem = blockIdx.x * EPB + w;        // grid sized exactly

    // ---- phase 0: stage W1 (vectorized), biases, z row ----
    for (int i4 = tid; i4 < DIN * DH / 4; i4 += blockDim.x) {
        int i = i4 * 4;
        *(float4*)&W1s[i / DH][i % DH] = ((const float4*)w1)[i4];
    }
    for (int k = tid; k < DH; k += blockDim.x) {
        b1s[k] = b1[k]; b2s[k] = b2[k]; w3s[k] = w3[k];
    }
    if (lane < DIN) zs[w][lane] = z[elem * DIN + lane];
    __syncthreads();

    // ---- phase 1: a1 -> h1, s1 ----
    #pragma unroll
    for (int q = 0; q < 4; ++q) {
        int j = lane + 32 * q;
        float a = b1s[j];
        #pragma unroll
        for (int i = 0; i < DIN; ++i) a += zs[w][i] * W1s[i][j];
        hs[w][j]  = softplus_f(a);
        s1s[w][j] = sigmoid_f(a);
    }
    __syncthreads();

    // ---- phase 2: a2 -> s2, g2 = s2*w3, d2 = w3*s2*(1-s2) ----
    if (useWs) {
        const float4* hv = (const float4*)hs[w];
        #pragma unroll
        for (int q = 0; q < 4; ++q) {
            int k = lane + 32 * q;
            const float4* w2r = (const float4*)(w2t + k * DH);   // row k of W2^T
            float a = b2s[k];
            for (int j4 = 0; j4 < DH / 4; ++j4) {
                float4 wv = w2r[j4];
                float4 h4 = hv[j4];
                a += wv.x * h4.x + wv.y * h4.y + wv.z * h4.z + wv.w * h4.w;
            }
            float s = sigmoid_f(a);
            g2s[w][k] = s * w3s[k];
            d2s[w][k] = w3s[k] * s * (1.0f - s);
        }
    } else {
        #pragma unroll
        for (int q = 0; q < 4; ++q) {
            int k = lane + 32 * q;
            float a = b2s[k];
            for (int j = 0; j < DH; ++j) a += hs[w][j] * w2[j * DH + k];
            float s = sigmoid_f(a);
            g2s[w][k] = s * w3s[k];
            d2s[w][k] = w3s[k] * s * (1.0f - s);
        }
    }
    __syncthreads();

    // ---- phase 3: v1 = W2 g2 ; d1 = s1(1-s1) v1 ; g1 = s1 v1 ----
    {
        const float4* gv = (const float4*)g2s[w];
        #pragma unroll
        for (int q = 0; q < 4; ++q) {
            int j = lane + 32 * q;
            const float4* w2r = (const float4*)(w2 + j * DH);    // row j of W2
            float v = 0.0f;
            for (int k4 = 0; k4 < DH / 4; ++k4) {
                float4 wv = w2r[k4];
                float4 g4 = gv[k4];
                v += wv.x * g4.x + wv.y * g4.y + wv.z * g4.z + wv.w * g4.w;
            }
            float s = s1s[w][j];
            d1s[w][j] = s * (1.0f - s) * v;
            g1s[w][j] = s * v;
        }
    }
    __syncthreads();

    // ---- phase 4: jac = W1 g1 (lanes 0..15) ----
    if (lane < DIN) {
        float acc = 0.0f;
        for (int j = 0; j < DH; ++j) acc += W1s[lane][j] * g1s[w][j];
        jacs[w][lane] = acc;
    }
    __syncthreads();

    // ---- phase 5: Q = (W1 .* s1) @ W2 via WMMA f32 16x16x4 ----
    const int hl   = lane & 15;
    const int koff = (lane >> 4) << 1;
    const int mrow = (lane >> 4) * 8;

    for (int t = 0; t < 8; ++t) {
        v8f acc = {0.f, 0.f, 0.f, 0.f, 0.f, 0.f, 0.f, 0.f};
        if (useWs) {
            for (int kk = 0; kk < DH; kk += 4) {
                int ka = kk + koff;
                v2f a, b;
                a.x = W1s[hl][ka]     * s1s[w][ka];
                a.y = W1s[hl][ka + 1] * s1s[w][ka + 1];
                float2 bb = w2f[(((kk >> 2) * 8) + t) * 32 + lane]; // 1x b64, coalesced
                b.x = bb.x; b.y = bb.y;
                acc = wmma4(a, b, acc);
            }
        } else {
            for (int kk = 0; kk < DH; kk += 4) {
                int ka = kk + koff;
                v2f a, b;
                a.x = W1s[hl][ka]     * s1s[w][ka];
                a.y = W1s[hl][ka + 1] * s1s[w][ka + 1];
                b.x = w2[(ka)     * DH + t * 16 + hl];
                b.y = w2[(ka + 1) * DH + t * 16 + hl];
                acc = wmma4(a, b, acc);
            }
        }
        #pragma unroll
        for (int r = 0; r < 8; ++r)
            Qs[w][mrow + r][t * 16 + hl] = acc[r];
    }
    __syncthreads();

    // ---- phase 6: H = (W1.*d1)@W1^T + (Q.*d2)@Q^T (one 16x16 tile) ----
    {
        v8f acc = {0.f, 0.f, 0.f, 0.f, 0.f, 0.f, 0.f, 0.f};
        for (int kk = 0; kk < DH; kk += 4) {
            int ka = kk + koff;
            v2f a, b;
            b.x = W1s[hl][ka];  b.y = W1s[hl][ka + 1];      // B[k][n] = W1[n][k]
            a.x = b.x * d1s[w][ka];
            a.y = b.y * d1s[w][ka + 1];
            acc = wmma4(a, b, acc);
        }
        for (int kk = 0; kk < DH; kk += 4) {
            int ka = kk + koff;
            v2f a, b;
            b.x = Qs[w][hl][ka];  b.y = Qs[w][hl][ka + 1];
            a.x = b.x * d2s[w][ka];
            a.y = b.y * d2s[w][ka + 1];
            acc = wmma4(a, b, acc);
        }
        if (lane >= 16) {
            #pragma unroll
            for (int r = 0; r < 8; ++r) Hs[w][r][hl] = acc[r];
        }
    }
    __syncthreads();

    // ---- phase 7: rhs = dL/dq - H[qdot,q]·qdot ; 8x8 partial-pivot solve ----
    if (lane == 0) {
        for (int i = 0; i < NDOF; ++i) {
            float r = jacs[w][i];
            for (int j = 0; j < NDOF; ++j) r -= Hs[w][i][j] * zs[w][NDOF + j];
            for (int j = 0; j < NDOF; ++j) Aug[w][i][j] = Hs[w][i][NDOF + j];
            Aug[w][i][NDOF] = r;
        }
        for (int p = 0; p < NDOF; ++p) {
            int pi = p; float pv = fabsf(Aug[w][p][p]);
            for (int i = p + 1; i < NDOF; ++i) {
                float v = fabsf(Aug[w][i][p]);
                if (v > pv) { pv = v; pi = i; }
            }
            if (pi != p)
                for (int j = p; j <= NDOF; ++j) {
                    float t = Aug[w][p][j]; Aug[w][p][j] = Aug[w][pi][j]; Aug[w][pi][j] = t;
                }
            float inv = 1.0f / Aug[w][p][p];
            for (int i = p + 1; i < NDOF; ++i) {
                float f = Aug[w][i][p] * inv;
                for (int j = p; j <= NDOF; ++j) Aug[w][i][j] -= f * Aug[w][p][j];
            }
        }
        for (int i = NDOF - 1; i >= 0; --i) {
            float s = Aug[w][i][NDOF];
            for (int j = i + 1; j < NDOF; ++j) s -= Aug[w][i][j] * Aug[w][j][NDOF];
            Aug[w][i][NDOF] = s / Aug[w][i][i];
        }
        for (int i = 0; i < NDOF; ++i) out[elem * NDOF + i] = Aug[w][i][NDOF];
    }
}

extern "C" void kernel_launch(void* const* d_in, const int* in_sizes, int n_in,
                              void* d_out, int out_size, void* d_ws, size_t ws_size,
                              hipStream_t stream) {
    (void)in_sizes; (void)n_in; (void)out_size;
    const float* z  = (const float*)d_in[0];
    const float* w1 = (const float*)d_in[1];
    const float* b1 = (const float*)d_in[2];
    const float* w2 = (const float*)d_in[3];
    const float* b2 = (const float*)d_in[4];
    const float* w3 = (const float*)d_in[5];
    // d_in[6] = b3: additive constant, no effect on grad/hessian
    float* out = (float*)d_out;

    const size_t W2F_BYTES = 8192 * sizeof(float2);   // 64 KB
    const size_t W2T_BYTES = DH * DH * sizeof(float); // 64 KB
    int useWs = (d_ws != nullptr && ws_size >= W2F_BYTES + W2T_BYTES) ? 1 : 0;

    float2* w2f = (float2*)d_ws;
    float*  w2t = (float*)((char*)d_ws + W2F_BYTES);

    if (useWs) {
        lnn_prep<<<dim3(64), dim3(256), 0, stream>>>(w2, w2f, w2t);
    }
    dim3 grid(BATCH / EPB), block(EPB * 32);
    lnn_kernel<<<grid, block, 0, stream>>>(z, w1, b1, w2, b2, w3,
                                           w2f, w2t, useWs, out);
}